// CLKG_compgatv3_convE_72404558676347
// MI455X (gfx1250) — compile-verified
//
#include <hip/hip_runtime.h>
#include <hip/hip_bf16.h>

// ---------------------------------------------------------------------------
// CompGATv3 (2 layers) + ConvE decoder for MI455X (gfx1250, wave32, WMMA).
// fp32 WMMA (V_WMMA_F32_16X16X4_F32) everywhere; 32x32 output tile per wave
// (4 accumulators) for 2x better bytes/FLOP; optional split-K for small-M
// GEMMs. Branch-free inner loops: OOB lanes clamp addresses (WMMA rows/cols
// never mix), pointer-increment addressing, 8B vector A-fragment loads.
// ---------------------------------------------------------------------------

#define N_ENT 50000
#define N_REL 500
#define DD    200
#define NE    200000
#define BB    256
#define NUM_IN 10368   // 32 * 18 * 18

typedef __attribute__((ext_vector_type(2))) float v2f;
typedef __attribute__((ext_vector_type(8))) float v8f;

#define WMMA_F32(a, b, c) \
    __builtin_amdgcn_wmma_f32_16x16x4_f32(false, (a), false, (b), (short)0, (c), false, false)

// ---------------------------------------------------------------------------
// Generic fp32 WMMA GEMM:  C[M,N] (+)= A[M,K] @ B  (+ bias[col])
// B element (k,n) at B[brs*k + bcs*n]  -> supports B and B^T.
// One wave per 32x32 output tile. Split-K via gridDim.z (kc = K-chunk,
// multiple of 4): chunks accumulate with atomicAdd into zeroed C and the
// z==0 chunk contributes the bias once.
// ---------------------------------------------------------------------------
__global__ __launch_bounds__(32)
void CLKG_gemm_wmma_f32(const float* __restrict__ A, const float* __restrict__ B,
                        float* __restrict__ C, int M, int N, int K,
                        int lda, long brs, long bcs, int ldc,
                        const float* __restrict__ bias, int kc, int accum)
{
    const int lane = threadIdx.x;
    const int row0 = blockIdx.x * 32;
    const int col0 = blockIdx.y * 32;
    const int k0   = blockIdx.z * kc;
    const int kend = min(K, k0 + kc);

    const int l15 = lane & 15;
    // clamped (always in-bounds) load coordinates; stores are guarded below
    const int arow0 = min(row0 + l15,      M - 1);
    const int arow1 = min(row0 + 16 + l15, M - 1);
    const int bcol0 = min(col0 + l15,      N - 1);
    const int bcol1 = min(col0 + 16 + l15, N - 1);
    const int kA = (lane >> 4) * 2;     // A: half-wave K pair select
    const int kB = (lane >> 4);         // B: half-wave K row select

    const v2f*   pa0  = (const v2f*)(A + (long)arow0 * lda + k0 + kA);
    const v2f*   pa1  = (const v2f*)(A + (long)arow1 * lda + k0 + kA);
    const float* pb0  = B + brs * (k0 + kB) + bcs * bcol0;
    const float* pb1  = B + brs * (k0 + kB) + bcs * bcol1;
    const long   b2   = brs * 2;
    const long   bstep = brs * 4;

    v8f acc00 = {}, acc01 = {}, acc10 = {}, acc11 = {};
#pragma unroll 2
    for (int k = k0; k < kend; k += 4) {
        v2f a0 = *pa0;
        v2f a1 = *pa1;
        v2f b0, b1;
        b0.x = pb0[0]; b0.y = pb0[b2];
        b1.x = pb1[0]; b1.y = pb1[b2];
        acc00 = WMMA_F32(a0, b0, acc00);
        acc01 = WMMA_F32(a0, b1, acc01);
        acc10 = WMMA_F32(a1, b0, acc10);
        acc11 = WMMA_F32(a1, b1, acc11);
        pa0 += 2;  pa1 += 2;          // 4 floats
        pb0 += bstep;  pb1 += bstep;
    }

    const int rb = (lane >> 4) * 8;
#pragma unroll
    for (int half = 0; half < 2; half++) {
        const int c = col0 + half * 16 + l15;
        if (c >= N) continue;
        const float bv = (bias && blockIdx.z == 0) ? bias[c] : 0.f;
        const v8f accT = half ? acc01 : acc00;   // rows row0..row0+15
        const v8f accB = half ? acc11 : acc10;   // rows row0+16..row0+31
#pragma unroll
        for (int v = 0; v < 8; v++) {
            const int r = row0 + rb + v;
            if (r < M) {
                const float val = accT[v] + bv;
                if (accum) atomicAdd(&C[(long)r * ldc + c], val);
                else       C[(long)r * ldc + c] = val;
            }
            const int r2 = r + 16;
            if (r2 < M) {
                const float val = accB[v] + bv;
                if (accum) atomicAdd(&C[(long)r2 * ldc + c], val);
                else       C[(long)r2 * ldc + c] = val;
            }
        }
    }
}

// ---------------------------------------------------------------------------
// Fused edge-message kernel: for a block of 16 edges,
//   comp = x[src] * rel[etype]        (staged in LDS)
//   msg  = comp @ W                   (13 waves, one 16x16 N-tile each, WMMA)
//   e    = leakyrelu(msg + hd[dst], 0.2) . a   (fused shuffle+LDS reduction)
// ---------------------------------------------------------------------------
__global__ __launch_bounds__(416)
void CLKG_edge_msg(const float* __restrict__ x, const float* __restrict__ relf,
                   const int* __restrict__ eidx,   // [2,E]: src then dst
                   const int* __restrict__ etype,
                   const float* __restrict__ W, const float* __restrict__ avec,
                   const float* __restrict__ hd,
                   float* __restrict__ msg, float* __restrict__ eout)
{
    __shared__ float compA[16 * DD];
    __shared__ float epart[16];
    __shared__ int   dsts[16];

    const int  tid  = threadIdx.x;
    const long eblk = (long)blockIdx.x * 16;

    for (int idx = tid; idx < 16 * DD; idx += 416) {
        const int  row = idx / DD;
        const int  d   = idx - row * DD;
        const long e   = eblk + row;
        const int  s   = eidx[e];
        const int  et  = etype[e];
        compA[idx] = x[(long)s * DD + d] * relf[(long)et * DD + d];
    }
    if (tid < 16) {
        epart[tid] = 0.f;
        dsts[tid]  = eidx[(long)NE + eblk + tid];
    }
    __syncthreads();

    const int  wave   = tid >> 5;
    const int  lane   = tid & 31;
    const int  col0   = wave * 16;
    const int  col    = col0 + (lane & 15);
    const bool cvalid = col < DD;
    const int  colc   = min(col, DD - 1);    // clamped load column
    const int  kA     = (lane >> 4) * 2;
    const int  kB     = (lane >> 4);
    const int  arow   = (lane & 15);

    const float* pa  = compA + arow * DD + kA;   // LDS, 8B-aligned pairs
    const float* pb0 = W + kB * DD + colc;
    const float* pb1 = pb0 + 2 * DD;

    v8f acc = {};
#pragma unroll 2
    for (int k = 0; k < DD; k += 4) {
        v2f a = *(const v2f*)pa;
        v2f b;
        b.x = *pb0;
        b.y = *pb1;
        acc = WMMA_F32(a, b, acc);
        pa  += 4;
        pb0 += 4 * DD;
        pb1 += 4 * DD;
    }

    const float aval  = cvalid ? avec[colc] : 0.f;   // zero kills masked cols
    const int   rbase = (lane >> 4) * 8;
#pragma unroll
    for (int v = 0; v < 8; v++) {
        const int  row = rbase + v;
        const long e   = eblk + row;
        const float mv = acc[v];
        if (cvalid) msg[e * DD + col] = mv;
        float t = mv + hd[(long)dsts[row] * DD + colc];
        t = fmaxf(t, 0.2f * t);                      // leaky_relu(t, 0.2)
        float p = t * aval;
        for (int o = 8; o > 0; o >>= 1) p += __shfl_xor(p, o, 16);
        if ((lane & 15) == 0) atomicAdd(&epart[row], p);
    }
    __syncthreads();
    if (tid < 16) eout[eblk + tid] = epart[tid];
}

// ---------------------------------------------------------------------------
// Segment softmax helpers (monotone float->uint map for atomicMax)
// ---------------------------------------------------------------------------
__global__ void CLKG_zero_u32(unsigned* p, long n)
{
    long i = (long)blockIdx.x * 256 + threadIdx.x;
    if (i < n) p[i] = 0u;
}

__global__ void CLKG_seg_max(const float* __restrict__ e, const int* __restrict__ eidx,
                             unsigned* __restrict__ mu)
{
    int i = blockIdx.x * 256 + threadIdx.x;
    if (i >= NE) return;
    const int d = eidx[NE + i];
    unsigned b = __float_as_uint(e[i]);
    unsigned u = (b & 0x80000000u) ? ~b : (b | 0x80000000u);
    atomicMax(&mu[d], u);
}

__global__ void CLKG_seg_expsum(float* __restrict__ e, const int* __restrict__ eidx,
                                const unsigned* __restrict__ mu, float* __restrict__ den)
{
    int i = blockIdx.x * 256 + threadIdx.x;
    if (i >= NE) return;
    const int d = eidx[NE + i];
    const unsigned u = mu[d];
    const unsigned b = (u & 0x80000000u) ? (u ^ 0x80000000u) : ~u;
    const float m  = __uint_as_float(b);
    const float ex = expf(e[i] - m);
    e[i] = ex;
    atomicAdd(&den[d], ex);
}

__global__ void CLKG_alpha(const float* __restrict__ ex, const int* __restrict__ eidx,
                           const float* __restrict__ den,
                           const float* __restrict__ alpha_prev,
                           float* __restrict__ alpha)
{
    int i = blockIdx.x * 256 + threadIdx.x;
    if (i >= NE) return;
    const int d = eidx[NE + i];
    float a = ex[i] / (den[d] + 1e-16f);
    if (alpha_prev) a = 0.5f * alpha_prev[i] + 0.5f * a;   // BETA = 0.5
    alpha[i] = a;
}

__global__ void CLKG_scatter_out(const float* __restrict__ msg,
                                 const float* __restrict__ alpha,
                                 const int* __restrict__ eidx,
                                 float* __restrict__ out)
{
    long i = (long)blockIdx.x * 256 + threadIdx.x;
    if (i >= (long)NE * DD) return;
    const int e = (int)(i / DD);
    const int d = (int)(i - (long)e * DD);
    const int ds = eidx[NE + e];
    atomicAdd(&out[(long)ds * DD + d], alpha[e] * msg[i]);
}

__global__ void CLKG_tanh_bias(float* __restrict__ out, const float* __restrict__ bvec, long n)
{
    long i = (long)blockIdx.x * 256 + threadIdx.x;
    if (i >= n) return;
    const int d = (int)(i % DD);
    out[i] = tanhf(out[i] + bvec[d]);
}

// ---------------------------------------------------------------------------
// ConvE decoder: stack/reshape -> bn0 -> 3x3 conv(32) -> bn1 -> PReLU -> flat
// One block per sample; 20x20 image + 32x3x3 weights live in LDS.
// ---------------------------------------------------------------------------
__global__ __launch_bounds__(256)
void CLKG_conv_decoder(const float* __restrict__ ent2, const float* __restrict__ rel2,
                       const int* __restrict__ hidx, const int* __restrict__ ridx,
                       const float* __restrict__ convw,
                       const float* __restrict__ bn0g, const float* __restrict__ bn0b,
                       const float* __restrict__ bn1g, const float* __restrict__ bn1b,
                       const float* __restrict__ prelu1, float* __restrict__ xflat)
{
    __shared__ float img[400];
    __shared__ float wc[288];
    const int b   = blockIdx.x;
    const int tid = threadIdx.x;
    const float inv = rsqrtf(1.f + 1e-5f);
    const int hh = hidx[b];
    const int rr = ridx[b];
    if (tid < 400) {
        const int c = tid & 1, q = tid >> 1;      // interleave head/rel rows
        const float v = (c == 0) ? ent2[(long)hh * DD + q] : rel2[(long)rr * DD + q];
        img[tid] = v * (bn0g[0] * inv) + bn0b[0];
    }
    if (tid < 288) wc[tid] = convw[tid];
    __syncthreads();

    const float slope = prelu1[0];
    for (int p = tid; p < 324; p += 256) {
        const int i = p / 18, j = p - i * 18;
        float patch[9];
#pragma unroll
        for (int di = 0; di < 3; di++)
#pragma unroll
            for (int dj = 0; dj < 3; dj++)
                patch[di * 3 + dj] = img[(i + di) * 20 + (j + dj)];
        for (int c = 0; c < 32; c++) {
            float acc = 0.f;
#pragma unroll
            for (int t = 0; t < 9; t++) acc += patch[t] * wc[c * 9 + t];
            acc = acc * (bn1g[c] * inv) + bn1b[c];
            acc = acc > 0.f ? acc : slope * acc;
            xflat[(long)b * NUM_IN + c * 324 + p] = acc;
        }
    }
}

__global__ void CLKG_z_epilogue(float* __restrict__ z, const float* __restrict__ g,
                                const float* __restrict__ bb, const float* __restrict__ slope)
{
    int i = blockIdx.x * 256 + threadIdx.x;
    if (i >= BB * DD) return;
    const int d = i % DD;
    const float inv = rsqrtf(1.f + 1e-5f);
    float v = z[i] * (g[d] * inv) + bb[d];
    z[i] = v > 0.f ? v : slope[0] * v;
}

// ---------------------------------------------------------------------------
extern "C" void kernel_launch(void* const* d_in, const int* in_sizes, int n_in,
                              void* d_out, int out_size, void* d_ws, size_t ws_size,
                              hipStream_t stream)
{
    const int*   edge_index = (const int*)d_in[0];   // [2,E]
    const int*   edge_type  = (const int*)d_in[1];
    const int*   h_idx      = (const int*)d_in[2];
    const int*   r_idx      = (const int*)d_in[3];
    /* d_in[4] = t (unused in forward) */
    const float* ent_emb  = (const float*)d_in[5];
    const float* rel_emb  = (const float*)d_in[6];
    const float* W1       = (const float*)d_in[7];
    const float* Wself1   = (const float*)d_in[8];
    const float* a1       = (const float*)d_in[9];
    const float* b1       = (const float*)d_in[10];
    const float* Wrel1    = (const float*)d_in[11];
    const float* W2       = (const float*)d_in[12];
    const float* Wself2   = (const float*)d_in[13];
    const float* a2       = (const float*)d_in[14];
    const float* b2       = (const float*)d_in[15];
    const float* Wrel2    = (const float*)d_in[16];
    const float* conv_w   = (const float*)d_in[17];
    const float* proj_w   = (const float*)d_in[18];
    const float* proj_b   = (const float*)d_in[19];
    const float* bias_ent = (const float*)d_in[20];
    const float* bn0_g    = (const float*)d_in[21];
    const float* bn0_b    = (const float*)d_in[22];
    const float* bn1_g    = (const float*)d_in[23];
    const float* bn1_b    = (const float*)d_in[24];
    const float* bnp_g    = (const float*)d_in[25];
    const float* bnp_b    = (const float*)d_in[26];
    const float* prelu1   = (const float*)d_in[27];
    const float* prelu2   = (const float*)d_in[28];

    float* out = (float*)d_out;

    // ---- workspace carve-up (aligned to 256B) ----
    char* ws = (char*)d_ws;
    size_t off = 0;
    auto carve = [&](size_t bytes) -> char* {
        char* p = ws + off;
        off += (bytes + 255) & ~(size_t)255;
        return p;
    };
    float*    msg    = (float*)carve((size_t)NE * DD * 4);      // 160 MB
    float*    hdall  = (float*)carve((size_t)N_ENT * DD * 4);   //  40 MB
    float*    ent1   = (float*)carve((size_t)N_ENT * DD * 4);   //  40 MB
    float*    ent2   = (float*)carve((size_t)N_ENT * DD * 4);   //  40 MB
    float*    rel1   = (float*)carve((size_t)N_REL * DD * 4);
    float*    rel2   = (float*)carve((size_t)N_REL * DD * 4);
    float*    ebuf   = (float*)carve((size_t)NE * 4);
    float*    alpha1 = (float*)carve((size_t)NE * 4);
    float*    alpha2 = (float*)carve((size_t)NE * 4);
    unsigned* mu     = (unsigned*)carve((size_t)N_ENT * 4);
    float*    den    = (float*)carve((size_t)N_ENT * 4);
    float*    xflat  = (float*)carve((size_t)BB * NUM_IN * 4);  // 10.6 MB
    float*    zbuf   = (float*)carve((size_t)BB * DD * 4);

    const dim3 gEnt((N_ENT + 31) / 32, (DD + 31) / 32);
    const dim3 gRel((N_REL + 31) / 32, (DD + 31) / 32);

    auto layer = [&](const float* x_in, const float* rf_in,
                     const float* W, const float* Wself, const float* avec,
                     const float* bvec, const float* Wrel,
                     const float* alpha_prev, float* alpha_out,
                     float* x_out, float* rel_out) {
        // per-entity self term:  hd = x @ Wself
        CLKG_gemm_wmma_f32<<<gEnt, 32, 0, stream>>>(x_in, Wself, hdall,
            N_ENT, DD, DD, DD, (long)DD, 1L, DD, nullptr, DD, 0);
        // relation transform:   rel_out = rel @ Wrel
        CLKG_gemm_wmma_f32<<<gRel, 32, 0, stream>>>(rf_in, Wrel, rel_out,
            N_REL, DD, DD, DD, (long)DD, 1L, DD, nullptr, DD, 0);
        // zero accumulators
        CLKG_zero_u32<<<(N_ENT + 255) / 256, 256, 0, stream>>>(mu, N_ENT);
        CLKG_zero_u32<<<(N_ENT + 255) / 256, 256, 0, stream>>>((unsigned*)den, N_ENT);
        CLKG_zero_u32<<<(int)(((long)N_ENT * DD + 255) / 256), 256, 0, stream>>>(
            (unsigned*)x_out, (long)N_ENT * DD);
        // fused composed-message WMMA + attention logits
        CLKG_edge_msg<<<NE / 16, 416, 0, stream>>>(x_in, rf_in, edge_index,
            edge_type, W, avec, hdall, msg, ebuf);
        // segment softmax over dst
        CLKG_seg_max<<<(NE + 255) / 256, 256, 0, stream>>>(ebuf, edge_index, mu);
        CLKG_seg_expsum<<<(NE + 255) / 256, 256, 0, stream>>>(ebuf, edge_index, mu, den);
        CLKG_alpha<<<(NE + 255) / 256, 256, 0, stream>>>(ebuf, edge_index, den,
            alpha_prev, alpha_out);
        // weighted segment sum + tanh(out + b)
        CLKG_scatter_out<<<(int)(((long)NE * DD + 255) / 256), 256, 0, stream>>>(
            msg, alpha_out, edge_index, x_out);
        CLKG_tanh_bias<<<(int)(((long)N_ENT * DD + 255) / 256), 256, 0, stream>>>(
            x_out, bvec, (long)N_ENT * DD);
    };

    // encoder
    layer(ent_emb, rel_emb, W1, Wself1, a1, b1, Wrel1, nullptr, alpha1, ent1, rel1);
    layer(ent1,    rel1,    W2, Wself2, a2, b2, Wrel2, alpha1,  alpha2, ent2, rel2);

    // ConvE decoder
    CLKG_conv_decoder<<<BB, 256, 0, stream>>>(ent2, rel2, h_idx, r_idx, conv_w,
        bn0_g, bn0_b, bn1_g, bn1_b, prelu1, xflat);

    // z = xflat @ proj_w + proj_b  (split-K x8 for parallelism; zbuf zeroed)
    CLKG_zero_u32<<<(BB * DD + 255) / 256, 256, 0, stream>>>((unsigned*)zbuf, BB * DD);
    CLKG_gemm_wmma_f32<<<dim3(BB / 32, (DD + 31) / 32, 8), 32, 0, stream>>>(
        xflat, proj_w, zbuf, BB, DD, NUM_IN, NUM_IN, (long)DD, 1L, DD,
        proj_b, NUM_IN / 8, 1);
    // BN + PReLU
    CLKG_z_epilogue<<<(BB * DD + 255) / 256, 256, 0, stream>>>(zbuf, bnp_g, bnp_b, prelu2);

    // scores = z @ ent2^T + bias_ent   (B strides: brs=1, bcs=DD)
    CLKG_gemm_wmma_f32<<<dim3(BB / 32, (N_ENT + 31) / 32, 1), 32, 0, stream>>>(
        zbuf, ent2, out, BB, N_ENT, DD, DD, 1L, (long)DD, N_ENT, bias_ent, DD, 0);
}